// InstanceLayer_68375879352930
// MI455X (gfx1250) — compile-verified
//
#include <hip/hip_runtime.h>
#include <math.h>

#define NCLS 80
#define NOUT 85
#define KCAND 300
#define MAXDET 25
#define CONF_T 0.25f
#define IOU_T 0.7f
#define MAXWH 7680.0f
#define BATCH 8

typedef float v2f __attribute__((ext_vector_type(2)));
typedef float v8f __attribute__((ext_vector_type(8)));

__constant__ float c_anchors[3][3][2] = {
  {{1.25f, 1.625f}, {2.0f, 3.75f}, {4.125f, 2.875f}},
  {{1.875f, 3.8125f}, {3.875f, 2.8125f}, {3.6875f, 7.4375f}},
  {{3.625f, 2.8125f}, {4.875f, 6.1875f}, {11.65625f, 10.1875f}}
};

__device__ __forceinline__ float sigmoidf_(float v) { return 1.0f / (1.0f + expf(-v)); }

// ---------------------------------------------------------------------------
// Kernel 1: decode. One thread per candidate (b, a, gy, gx).
// Only computes what NMS needs: box xywh, score (=conf if valid else 0), class.
// ---------------------------------------------------------------------------
__global__ void decode_kernel(const float* __restrict__ x, int lvl, int ny, int nx,
                              float* __restrict__ cbox, float* __restrict__ cscore,
                              int* __restrict__ ccls) {
  int N = 3 * ny * nx;
  int idx = blockIdx.x * blockDim.x + threadIdx.x;
  if (idx >= BATCH * N) return;
  int b = idx / N;
  int r = idx - b * N;
  int hw = ny * nx;
  int a = r / hw;
  int g = r - a * hw;
  int gy = g / nx, gx = g - gy * nx;
  const float* xp = x + (size_t)idx * NOUT;
  float sx = sigmoidf_(xp[0]);
  float sy = sigmoidf_(xp[1]);
  float sw = sigmoidf_(xp[2]);
  float sh = sigmoidf_(xp[3]);
  float obj = sigmoidf_(xp[4]);
  // argmax over raw class logits == argmax over sigmoid(logits) (monotone)
  float mx = xp[5];
  int mc = 0;
#pragma unroll 4
  for (int c = 1; c < NCLS; c++) {
    float v = xp[5 + c];
    if (v > mx) { mx = v; mc = c; }
  }
  float conf = sigmoidf_(mx) * obj;
  float cx = sx * 2.0f + ((float)gx - 0.5f);
  float cy = sy * 2.0f + ((float)gy - 0.5f);
  float tw = sw * 2.0f, th = sh * 2.0f;
  float w = tw * tw * c_anchors[lvl][a][0];
  float h = th * th * c_anchors[lvl][a][1];
  bool valid = (obj > CONF_T) && (conf > CONF_T);
  float* bp = cbox + (size_t)idx * 4;
  bp[0] = cx; bp[1] = cy; bp[2] = w; bp[3] = h;
  cscore[idx] = valid ? conf : 0.0f;
  ccls[idx] = mc;
}

// ---------------------------------------------------------------------------
// Kernel 2: top-K per image. Scores staged in (big, 320KB-capable) LDS,
// 300 iterative argmax reductions (value desc, index asc tie-break == lax.top_k).
// ---------------------------------------------------------------------------
__global__ void topk_kernel(const float* __restrict__ cscore, int N,
                            float* __restrict__ topv, int* __restrict__ topi) {
  extern __shared__ float s[];   // N floats
  __shared__ float rv[1024];
  __shared__ int ri[1024];
  int b = blockIdx.x;
  int tid = threadIdx.x;
  int nt = blockDim.x;
  for (int i = tid; i < N; i += nt) s[i] = cscore[(size_t)b * N + i];
  __syncthreads();
  for (int k = 0; k < KCAND; k++) {
    float bv = -2.0f;
    int bi = 0x7fffffff;
    for (int i = tid; i < N; i += nt) {
      float v = s[i];
      if (v > bv || (v == bv && i < bi)) { bv = v; bi = i; }
    }
    rv[tid] = bv; ri[tid] = bi;
    __syncthreads();
    for (int off = nt >> 1; off > 0; off >>= 1) {
      if (tid < off) {
        float v2 = rv[tid + off]; int i2 = ri[tid + off];
        if (v2 > rv[tid] || (v2 == rv[tid] && i2 < ri[tid])) { rv[tid] = v2; ri[tid] = i2; }
      }
      __syncthreads();
    }
    if (tid == 0) {
      topv[b * KCAND + k] = rv[0];
      topi[b * KCAND + k] = ri[0];
      s[ri[0]] = -1.0f;   // exclude from further rounds
    }
    __syncthreads();
  }
}

// ---------------------------------------------------------------------------
// Kernel 3: sequential NMS per image (one block = 10 waves, barrier per step,
// matches lax.fori_loop semantics exactly). Scatters up to 25 kept boxes.
// ---------------------------------------------------------------------------
__global__ void nms_kernel(const float* __restrict__ cbox, const int* __restrict__ ccls,
                           const float* __restrict__ topv, const int* __restrict__ topi,
                           int N, float* __restrict__ obox, int* __restrict__ ovalid) {
  __shared__ float bx1[KCAND], by1[KCAND], bx2[KCAND], by2[KCAND], boff[KCAND];
  __shared__ int vc[KCAND], supp[KCAND], keep[KCAND];
  int b = blockIdx.x, j = threadIdx.x;
  if (j < MAXDET) ovalid[b * MAXDET + j] = 0;
  if (j < MAXDET * 4) obox[b * MAXDET * 4 + j] = 0.0f;
  if (j < KCAND) {
    int id = topi[b * KCAND + j];
    float tv = topv[b * KCAND + j];
    const float* bp = cbox + ((size_t)b * N + id) * 4;
    float cx = bp[0], cy = bp[1], w = bp[2], h = bp[3];
    float off = (float)ccls[(size_t)b * N + id] * MAXWH;
    bx1[j] = cx - 0.5f * w + off;
    by1[j] = cy - 0.5f * h + off;
    bx2[j] = cx + 0.5f * w + off;
    by2[j] = cy + 0.5f * h + off;
    boff[j] = off;
    vc[j] = (tv > 0.0f) ? 1 : 0;
    supp[j] = 0;
  }
  __syncthreads();
  for (int i = 0; i < KCAND; i++) {
    int keep_i = vc[i] && !supp[i];
    if (keep_i && j < KCAND && j > i) {
      float ltx = fmaxf(bx1[i], bx1[j]);
      float lty = fmaxf(by1[i], by1[j]);
      float rbx = fminf(bx2[i], bx2[j]);
      float rby = fminf(by2[i], by2[j]);
      float iw = fmaxf(rbx - ltx, 0.0f), ih = fmaxf(rby - lty, 0.0f);
      float inter = iw * ih;
      float a1 = (bx2[i] - bx1[i]) * (by2[i] - by1[i]);
      float a2 = (bx2[j] - bx1[j]) * (by2[j] - by1[j]);
      float iou = inter / (a1 + a2 - inter + 1e-7f);
      if (iou > IOU_T) supp[j] = 1;
    }
    __syncthreads();
  }
  if (j < KCAND) keep[j] = vc[j] && !supp[j];
  __syncthreads();
  if (j < KCAND && keep[j]) {
    int rank = 0;
    for (int t = 0; t < j; t++) rank += keep[t];
    if (rank < MAXDET) {
      float off = boff[j];
      obox[(b * MAXDET + rank) * 4 + 0] = bx1[j] - off;
      obox[(b * MAXDET + rank) * 4 + 1] = by1[j] - off;
      obox[(b * MAXDET + rank) * 4 + 2] = bx2[j] - off;
      obox[(b * MAXDET + rank) * 4 + 3] = by2[j] - off;
      ovalid[b * MAXDET + rank] = 1;
    }
  }
}

// ---------------------------------------------------------------------------
// Kernel 4: fused ROI-align (7x7, ratio 2) + 1x1 conv (C->84) via f32 WMMA.
// One block per (image, box); 8 waves; 16-channel LDS tiles; accumulators
// (6 Mtiles x 4 Ntiles = 24 tiles, 3 per wave) live across the channel loop.
// ---------------------------------------------------------------------------
#define CT 16
__global__ void roiconv_kernel(const float* __restrict__ feat, const float* __restrict__ cw,
                               const float* __restrict__ cb, const float* __restrict__ obox,
                               int C, int H, int W, float* __restrict__ hflat) {
  __shared__ int s_ylo[14], s_yhi[14], s_xlo[14], s_xhi[14];
  __shared__ float s_fy[14], s_my[14], s_fx[14], s_mx[14];
  __shared__ float roi[CT][52];   // [channel within tile][spatial 0..48]
  int blk = blockIdx.x;           // b*25 + n
  int bimg = blk / 25;
  int tid = threadIdx.x;
  if (tid < 28) {
    int axis = tid / 14;          // 0: y, 1: x
    int t = tid - axis * 14;
    float lo = obox[blk * 4 + (axis == 0 ? 1 : 0)];
    float hi = obox[blk * 4 + (axis == 0 ? 3 : 2)];
    int L = (axis == 0) ? H : W;
    float r = fmaxf(hi - lo, 1.0f);
    float cc = lo + ((float)t + 0.5f) / 14.0f * r;
    float m = (cc >= -1.0f && cc <= (float)L) ? 1.0f : 0.0f;
    cc = fmaxf(cc, 0.0f);
    int l = (int)floorf(cc);
    l = min(max(l, 0), L - 1);
    int hh = min(l + 1, L - 1);
    float f = fminf(fmaxf(cc - (float)l, 0.0f), 1.0f);
    if (axis == 0) { s_ylo[t] = l; s_yhi[t] = hh; s_fy[t] = f; s_my[t] = m; }
    else           { s_xlo[t] = l; s_xhi[t] = hh; s_fx[t] = f; s_mx[t] = m; }
  }
  __syncthreads();
  int lane = tid & 31;
  int wave = tid >> 5;
  int kh = lane >> 4;
  int ln = lane & 15;
  v8f acc[3] = {};
  const float* fb = feat + (size_t)bimg * C * H * W;
  for (int c0 = 0; c0 < C; c0 += CT) {
    // stage 16x49 roi tile in LDS
    for (int t = tid; t < CT * 49; t += 256) {
      int lc = t / 49;
      int p = t - lc * 49;
      int i = p / 7, jx = p - i * 7;
      const float* fc = fb + (size_t)(c0 + lc) * H * W;
      float a4 = 0.0f;
      for (int sy = 0; sy < 2; sy++) {
        int syi = 2 * i + sy;
        int ylo = s_ylo[syi], yhi = s_yhi[syi];
        float fy = s_fy[syi], my = s_my[syi];
        for (int sx = 0; sx < 2; sx++) {
          int sxi = 2 * jx + sx;
          int xlo = s_xlo[sxi], xhi = s_xhi[sxi];
          float fx = s_fx[sxi], mxm = s_mx[sxi];
          float v00 = fc[ylo * W + xlo], v01 = fc[ylo * W + xhi];
          float v10 = fc[yhi * W + xlo], v11 = fc[yhi * W + xhi];
          float v = v00 * (1.0f - fy) * (1.0f - fx) + v01 * (1.0f - fy) * fx +
                    v10 * fy * (1.0f - fx) + v11 * fy * fx;
          a4 += v * (my * mxm);
        }
      }
      roi[lc][p] = a4 * 0.25f;
    }
    __syncthreads();
    // WMMA: out[o][p] += cw[o][c] * roi[c][p]; 4 k-steps of 4 per 16-ch tile
    for (int s = 0; s < 3; s++) {
      int pair = wave * 3 + s;
      int mt = pair >> 2;
      int nt = pair & 3;
      int o = mt * 16 + ln;
      int n = nt * 16 + ln;
      for (int kk = 0; kk < 4; kk++) {
        int kidx = kk * 4 + kh * 2;   // A: lanes0-15 K={0,1}, lanes16-31 K={2,3}
        v2f a, bb;
        a.x = (o < 84) ? cw[(size_t)o * C + c0 + kidx] : 0.0f;
        a.y = (o < 84) ? cw[(size_t)o * C + c0 + kidx + 1] : 0.0f;
        bb.x = (n < 49) ? roi[kidx][n] : 0.0f;
        bb.y = (n < 49) ? roi[kidx + 1][n] : 0.0f;
        acc[s] = __builtin_amdgcn_wmma_f32_16x16x4_f32(
            false, a, false, bb, (short)0, acc[s], false, false);
      }
    }
    __syncthreads();
  }
  // epilogue: D layout — VGPR r: lanes0-15 M=r, lanes16-31 M=r+8
  int mh = kh * 8;
  for (int s = 0; s < 3; s++) {
    int pair = wave * 3 + s;
    int mt = pair >> 2;
    int nt = pair & 3;
    int n = nt * 16 + ln;
    for (int rr = 0; rr < 8; rr++) {
      int o = mt * 16 + mh + rr;
      if (o < 84 && n < 49)
        hflat[(size_t)blk * 4116 + o * 49 + n] = acc[s][rr] + cb[o];
    }
  }
}

// ---------------------------------------------------------------------------
// Kernel 5: generic f32 WMMA GEMM  C(M,N) = A(M,K) @ W(K,N) + bias [+ relu]
// One wave per 16x16 output tile; K stepped by 4 (v_wmma_f32_16x16x4_f32).
// ---------------------------------------------------------------------------
__global__ void wmma_gemm_kernel(const float* __restrict__ A, const float* __restrict__ Wt,
                                 const float* __restrict__ bias, float* __restrict__ Cm,
                                 int M, int N, int K, int relu) {
  int lane = threadIdx.x & 31;
  int mt = blockIdx.x, nt = blockIdx.y;
  int ln = lane & 15;
  int kh = lane >> 4;
  int row = mt * 16 + ln;
  int col = nt * 16 + ln;
  v8f acc = {};
  for (int k = 0; k < K; k += 4) {
    int ka = k + kh * 2;
    v2f a, bb;
    a.x = (row < M && ka < K) ? A[(size_t)row * K + ka] : 0.0f;
    a.y = (row < M && ka + 1 < K) ? A[(size_t)row * K + ka + 1] : 0.0f;
    bb.x = (col < N && ka < K) ? Wt[(size_t)ka * N + col] : 0.0f;
    bb.y = (col < N && ka + 1 < K) ? Wt[(size_t)(ka + 1) * N + col] : 0.0f;
    acc = __builtin_amdgcn_wmma_f32_16x16x4_f32(
        false, a, false, bb, (short)0, acc, false, false);
  }
  int mh = kh * 8;
  for (int rr = 0; rr < 8; rr++) {
    int m = mt * 16 + mh + rr;
    if (m < M && col < N) {
      float v = acc[rr] + bias[col];
      if (relu) v = fmaxf(v, 0.0f);
      Cm[(size_t)m * N + col] = v;
    }
  }
}

// ---------------------------------------------------------------------------
// Kernel 6: mask by valid, pad 25 -> 85 rows, write (B,85,1) slab of output.
// ---------------------------------------------------------------------------
__global__ void writeout_kernel(const float* __restrict__ m3, const int* __restrict__ ovalid,
                                float* __restrict__ out) {
  int t = blockIdx.x * blockDim.x + threadIdx.x;
  if (t >= BATCH * NOUT) return;
  int b = t / NOUT, r = t - b * NOUT;
  float v = 0.0f;
  if (r < MAXDET) v = m3[b * MAXDET + r] * (ovalid[b * MAXDET + r] ? 1.0f : 0.0f);
  out[t] = v;
}

// ---------------------------------------------------------------------------
extern "C" void kernel_launch(void* const* d_in, const int* in_sizes, int n_in,
                              void* d_out, int out_size, void* d_ws, size_t ws_size,
                              hipStream_t stream) {
  (void)in_sizes; (void)n_in; (void)out_size; (void)ws_size;
  static const int NYS[3] = {80, 40, 20}, NXS[3] = {80, 40, 20}, CS[3] = {256, 512, 1024};
  const int NMAX = 3 * 80 * 80;
  // workspace partition (floats); ~7.2 MB total, reused across levels
  float* ws = (float*)d_ws;
  size_t o = 0;
  float* cand_box = ws + o;   o += (size_t)BATCH * NMAX * 4;
  float* cand_score = ws + o; o += (size_t)BATCH * NMAX;
  int* cand_cls = (int*)(ws + o); o += (size_t)BATCH * NMAX;
  float* topv = ws + o;       o += (size_t)BATCH * KCAND;
  int* topi = (int*)(ws + o); o += (size_t)BATCH * KCAND;
  float* nbox = ws + o;       o += (size_t)BATCH * MAXDET * 4;
  int* nvalid = (int*)(ws + o); o += (size_t)BATCH * MAXDET;
  float* hflat = ws + o;      o += (size_t)BATCH * MAXDET * 4116;
  float* m1 = ws + o;         o += (size_t)BATCH * MAXDET * 100;
  float* m2 = ws + o;         o += (size_t)BATCH * MAXDET * 100;
  float* m3 = ws + o;         o += (size_t)BATCH * MAXDET;
  float* out = (float*)d_out;

  for (int lvl = 0; lvl < 3; lvl++) {
    const float* feat = (const float*)d_in[10 * lvl + 0];
    const float* x    = (const float*)d_in[10 * lvl + 1];
    const float* cwp  = (const float*)d_in[10 * lvl + 2];
    const float* cbp  = (const float*)d_in[10 * lvl + 3];
    const float* w1   = (const float*)d_in[10 * lvl + 4];
    const float* b1   = (const float*)d_in[10 * lvl + 5];
    const float* w2   = (const float*)d_in[10 * lvl + 6];
    const float* b2   = (const float*)d_in[10 * lvl + 7];
    const float* w3   = (const float*)d_in[10 * lvl + 8];
    const float* b3   = (const float*)d_in[10 * lvl + 9];
    int ny = NYS[lvl], nx = NXS[lvl], C = CS[lvl];
    int N = 3 * ny * nx;

    decode_kernel<<<(BATCH * N + 255) / 256, 256, 0, stream>>>(
        x, lvl, ny, nx, cand_box, cand_score, cand_cls);
    topk_kernel<<<BATCH, 1024, (size_t)N * sizeof(float), stream>>>(
        cand_score, N, topv, topi);
    nms_kernel<<<BATCH, 320, 0, stream>>>(
        cand_box, cand_cls, topv, topi, N, nbox, nvalid);
    roiconv_kernel<<<BATCH * MAXDET, 256, 0, stream>>>(
        feat, cwp, cbp, nbox, C, ny, nx, hflat);
    wmma_gemm_kernel<<<dim3(13, 7), 32, 0, stream>>>(hflat, w1, b1, m1, 200, 100, 4116, 1);
    wmma_gemm_kernel<<<dim3(13, 7), 32, 0, stream>>>(m1, w2, b2, m2, 200, 100, 100, 1);
    wmma_gemm_kernel<<<dim3(13, 1), 32, 0, stream>>>(m2, w3, b3, m3, 200, 1, 100, 0);
    writeout_kernel<<<(BATCH * NOUT + 255) / 256, 256, 0, stream>>>(
        m3, nvalid, out + lvl * BATCH * NOUT);
  }
}